// MaskedEdgeHistoryEncoder_90134183674252
// MI455X (gfx1250) — compile-verified
//
#include <hip/hip_runtime.h>
#include <math.h>

// ---------------- problem constants ----------------
#define BSZ   2048
#define SEQL  50
#define ATTD  64
#define HIDD  128
#define OUTD  64
#define NPAIR 8            // (b,s) pairs per attention block
#define EROWS (NPAIR * 9)  // 72 live edge rows (rows 72..79 are don't-care)
#define MTIL  5            // 80 rows = 5 M tiles
#define NTIL  4            // 64 cols = 4 N tiles

typedef float v2f __attribute__((ext_vector_type(2)));
typedef float v8f __attribute__((ext_vector_type(8)));

// fp32-exact WMMA: D(16x16,f32) = A(16x4,f32) * B(4x16,f32) + C
__device__ __forceinline__ v8f wmma4(v2f a, v2f b, v8f c) {
  return __builtin_amdgcn_wmma_f32_16x16x4_f32(
      /*neg_a=*/false, a, /*neg_b=*/false, b,
      /*c_mod=*/(short)0, c, /*reuse_a=*/false, /*reuse_b=*/false);
}

__device__ __forceinline__ float sigm(float x) { return 1.0f / (1.0f + __expf(-x)); }

// CDNA5 async global->LDS copy (GVS mode: SGPR64 base + VGPR32 offset),
// tracked by ASYNCcnt. Inline asm = portable across both toolchains.
__device__ __forceinline__ void async_load_b128(uint32_t lds_off, uint32_t goff,
                                                const float* base) {
  asm volatile("global_load_async_to_lds_b128 %0, %1, %2"
               :: "v"(lds_off), "v"(goff), "s"(base)
               : "memory");
}

__device__ __forceinline__ void wait_async0() {
#if __has_builtin(__builtin_amdgcn_s_wait_asynccnt)
  __builtin_amdgcn_s_wait_asynccnt(0);
#else
  asm volatile("s_wait_asynccnt 0x0" ::: "memory");
#endif
}

// =====================================================================
// Kernel 1: fused edge linear (19->64), Q/V (64->64), K (6->64),
// masked 9-way softmax, context gather. ctx written (SEQ, BS, 64).
// WMMA A operands come from LDS (Wq/Wv B operands from global) with no lane-divergent control flow.
// =====================================================================
__global__ __launch_bounds__(128) void attn_ctx_kernel(
    const float* __restrict__ seqs,    // (BS,SEQ,3,3,6)
    const float* __restrict__ etys,    // (BS,SEQ,3,3,4)
    const int*   __restrict__ masks,   // (BS,SEQ,3,3)
    const float* __restrict__ W_fus,   // (64,19)
    const float* __restrict__ b_fus,   // (64)
    const float* __restrict__ Wk,      // (64,6)
    const float* __restrict__ bk,      // (64)
    const float* __restrict__ Wq,      // (64,64)
    const float* __restrict__ bq,      // (64)
    const float* __restrict__ Wv,      // (64,64)
    const float* __restrict__ bv,      // (64)
    float* __restrict__ ctx)           // (SEQ,BS,64)
{
  __shared__ float sF  [MTIL * 16][20];   // padded features (rows>=72, col 19 = 0)
  __shared__ float sWf [ATTD][22];        // W_fus[n][k], col>=19 zero, padded stride
  __shared__ int   sMsk[NPAIR][9];
  __shared__ float sE  [MTIL * 16][66];   // edge_info (padded stride)
  __shared__ float sQV [MTIL * 16][66];   // q, later v
  __shared__ float sKey[NPAIR][ATTD];
  __shared__ float sAtt[NPAIR][4][9];

  const int tid  = threadIdx.x;
  const int wave = tid >> 5;
  const int lane = tid & 31;
  const int lm   = lane & 15;   // M (A) / N (B,C) index
  const int lh   = lane >> 4;   // K-pair half selector
  const int gp0  = blockIdx.x * NPAIR;   // first flat (b*SEQ+s) pair

  // ---- staging: padded feature matrix, padded W_fus copy, masks ----
  for (int i = tid; i < MTIL * 16 * 20; i += 128) {
    const int r = i / 20, k = i % 20;
    float f = 0.0f;
    if (r < EROWS) {
      const int p = r / 9, nn = r % 9;
      if (k < 6)       f = seqs[(size_t)(gp0 + p) * 54 + nn * 6 + k];
      else if (k < 10) f = etys[(size_t)(gp0 + p) * 36 + nn * 4 + (k - 6)];
      else if (k < 19) f = (k - 10 == nn) ? 1.0f : 0.0f;
    }
    sF[r][k] = f;
  }
  for (int i = tid; i < ATTD * 20; i += 128) {
    const int n = i / 20, k = i % 20;
    sWf[n][k] = (k < 19) ? W_fus[n * 19 + k] : 0.0f;
  }
  for (int i = tid; i < NPAIR * 9; i += 128)
    ((int*)sMsk)[i] = masks[(size_t)gp0 * 9 + i];
  __syncthreads();

  // ---- fusion GEMM: (80x20) x (20x64) -> sE ----
  for (int tile = wave; tile < MTIL * NTIL; tile += 4) {
    const int mt = tile / NTIL, nt = tile % NTIL;
    const int n  = nt * 16 + lm;
    const int r  = mt * 16 + lm;
    v8f acc;
    const float bias = b_fus[n];
#pragma unroll
    for (int i = 0; i < 8; ++i) acc[i] = bias;
#pragma unroll
    for (int k0 = 0; k0 < 20; k0 += 4) {
      const int ks = k0 + 2 * lh;
      v2f a, b;
      a.x = sF[r][ks];  a.y = sF[r][ks + 1];
      b.x = sWf[n][ks]; b.y = sWf[n][ks + 1];
      acc = wmma4(a, b, acc);
    }
#pragma unroll
    for (int i = 0; i < 8; ++i) sE[mt * 16 + lh * 8 + i][n] = acc[i];
  }

  // ---- key (6->64) per pair, plain VALU (tiny) ----
  for (int i = tid; i < NPAIR * ATTD; i += 128) {
    const int p = i / ATTD, j = i % ATTD;
    float acc = bk[j];
#pragma unroll
    for (int k = 0; k < 6; ++k) acc += sF[p * 9 + 4][k] * Wk[j * 6 + k];
    sKey[p][j] = acc;
  }
  __syncthreads();

  // ---- Q GEMM: sE(80x64) x Wq^T -> sQV ----
  for (int tile = wave; tile < MTIL * NTIL; tile += 4) {
    const int mt = tile / NTIL, nt = tile % NTIL;
    const int n  = nt * 16 + lm;
    const int r  = mt * 16 + lm;
    v8f acc;
    const float bias = bq[n];
#pragma unroll
    for (int i = 0; i < 8; ++i) acc[i] = bias;
#pragma unroll
    for (int k0 = 0; k0 < 64; k0 += 4) {
      const int ks = k0 + 2 * lh;
      v2f a, b;
      a.x = sE[r][ks];       a.y = sE[r][ks + 1];
      b.x = Wq[n * 64 + ks]; b.y = Wq[n * 64 + ks + 1];
      acc = wmma4(a, b, acc);
    }
#pragma unroll
    for (int i = 0; i < 8; ++i) sQV[mt * 16 + lh * 8 + i][n] = acc[i];
  }
  __syncthreads();

  // ---- masked softmax over 9 neighbors, per (pair, head) ----
  if (tid < NPAIR * 4) {
    const int p = tid >> 2, h = tid & 3;
    float sc[9], mx = -3.0e38f;
#pragma unroll
    for (int nn = 0; nn < 9; ++nn) {
      float s = 0.0f;
#pragma unroll
      for (int d = 0; d < 16; ++d) s += sKey[p][h * 16 + d] * sQV[p * 9 + nn][h * 16 + d];
      s *= 0.25f;                              // 1/sqrt(16)
      if (sMsk[p][nn] == 0) s = -1.0e10f;
      sc[nn] = s; mx = fmaxf(mx, s);
    }
    float sum = 0.0f;
#pragma unroll
    for (int nn = 0; nn < 9; ++nn) { sc[nn] = __expf(sc[nn] - mx); sum += sc[nn]; }
    const float inv = 1.0f / sum;
#pragma unroll
    for (int nn = 0; nn < 9; ++nn) sAtt[p][h][nn] = sc[nn] * inv;
  }
  __syncthreads();

  // ---- V GEMM: sE x Wv^T -> sQV (overwrites q) ----
  for (int tile = wave; tile < MTIL * NTIL; tile += 4) {
    const int mt = tile / NTIL, nt = tile % NTIL;
    const int n  = nt * 16 + lm;
    const int r  = mt * 16 + lm;
    v8f acc;
    const float bias = bv[n];
#pragma unroll
    for (int i = 0; i < 8; ++i) acc[i] = bias;
#pragma unroll
    for (int k0 = 0; k0 < 64; k0 += 4) {
      const int ks = k0 + 2 * lh;
      v2f a, b;
      a.x = sE[r][ks];       a.y = sE[r][ks + 1];
      b.x = Wv[n * 64 + ks]; b.y = Wv[n * 64 + ks + 1];
      acc = wmma4(a, b, acc);
    }
#pragma unroll
    for (int i = 0; i < 8; ++i) sQV[mt * 16 + lh * 8 + i][n] = acc[i];
  }
  __syncthreads();

  // ---- ctx[p][h*16+d] = sum_n att * v ; write (SEQ,BS,64) ----
  for (int i = tid; i < NPAIR * ATTD; i += 128) {
    const int p = i / ATTD, hd = i % ATTD, h = hd >> 4;
    float acc = 0.0f;
#pragma unroll
    for (int nn = 0; nn < 9; ++nn) acc += sAtt[p][h][nn] * sQV[p * 9 + nn][hd];
    const int gp = gp0 + p;
    const int b = gp / SEQL, s = gp % SEQL;
    ctx[((size_t)s * BSZ + b) * ATTD + hd] = acc;
  }
}

// =====================================================================
// Kernel 2: full LSTM (50 steps) + output projection, one launch.
// Each block owns 16 batch rows for the whole sequence: h lives in LDS,
// c lives in registers. x_{t+1} is prefetched into a double buffer with
// CDNA5 async global->LDS copies, hidden behind the gate GEMM.
// =====================================================================
__global__ __launch_bounds__(256) void lstm_out_kernel(
    const float* __restrict__ ctx,    // (SEQ,BS,64)
    const float* __restrict__ W_ih,   // (512,64)
    const float* __restrict__ W_hh,   // (512,128)
    const float* __restrict__ b_ih,   // (512)
    const float* __restrict__ b_hh,   // (512)
    const float* __restrict__ W_out,  // (64,128)
    const float* __restrict__ b_out,  // (64)
    float* __restrict__ out)          // (BS,64)
{
  __shared__ alignas(16) float sX[2][16][68];  // x double buffer, 16B-aligned rows
  __shared__ float sH[16][130];                // h state (padded stride)
  __shared__ float sG[16][4 * HIDD];           // gate accumulators

  const int tid  = threadIdx.x;
  const int wave = tid >> 5;
  const int lane = tid & 31;
  const int lm   = lane & 15;
  const int lh   = lane >> 4;
  const int row0 = blockIdx.x * 16;

  // per-thread async-copy slot: 16 bytes of the 16x64 x-tile
  const int xr  = tid >> 4;              // row 0..15
  const int xc  = (tid & 15) << 2;       // col (floats) 0..60
  const uint32_t goff = (uint32_t)(tid << 4);          // global byte offset in tile
  const uint32_t ldsA = (uint32_t)(size_t)&sX[0][xr][xc];
  const uint32_t ldsB = (uint32_t)(size_t)&sX[1][xr][xc];

  float creg[8];                        // c for (row = 2*ii + tid/128, col = tid%128)
#pragma unroll
  for (int ii = 0; ii < 8; ++ii) creg[ii] = 0.0f;
  for (int i = tid; i < 16 * 130; i += 256) (&sH[0][0])[i] = 0.0f;   // h0 = 0

  // preload x_0 into buffer 0
  async_load_b128(ldsA, goff, ctx + (size_t)row0 * ATTD);
  wait_async0();
  __syncthreads();

  for (int t = 0; t < SEQL; ++t) {
    // prefetch x_{t+1} into the other buffer (hidden behind the GEMM)
    if (t + 1 < SEQL)
      async_load_b128((t & 1) ? ldsA : ldsB, goff,
                      ctx + ((size_t)(t + 1) * BSZ + row0) * ATTD);
    const float (*xb)[68] = sX[t & 1];

    // gates = x@W_ih^T + h@W_hh^T + b : 32 N-tiles over 8 waves
    for (int nt = wave; nt < 32; nt += 8) {
      const int n = nt * 16 + lm;
      v8f acc;
      const float bias = b_ih[n] + b_hh[n];
#pragma unroll
      for (int i = 0; i < 8; ++i) acc[i] = bias;
#pragma unroll
      for (int k0 = 0; k0 < ATTD; k0 += 4) {
        const int ks = k0 + 2 * lh;
        v2f a, b;
        a.x = xb[lm][ks];          a.y = xb[lm][ks + 1];
        b.x = W_ih[n * ATTD + ks]; b.y = W_ih[n * ATTD + ks + 1];
        acc = wmma4(a, b, acc);
      }
#pragma unroll
      for (int k0 = 0; k0 < HIDD; k0 += 4) {
        const int ks = k0 + 2 * lh;
        v2f a, b;
        a.x = sH[lm][ks];          a.y = sH[lm][ks + 1];
        b.x = W_hh[n * HIDD + ks]; b.y = W_hh[n * HIDD + ks + 1];
        acc = wmma4(a, b, acc);
      }
#pragma unroll
      for (int i = 0; i < 8; ++i) sG[lh * 8 + i][n] = acc[i];
    }
    __syncthreads();

    // elementwise cell update: h -> LDS, c -> registers
#pragma unroll
    for (int ii = 0; ii < 8; ++ii) {
      const int i = tid + ii * 256;
      const int r = i >> 7, j = i & 127;
      const float ig = sigm(sG[r][j]);
      const float fg = sigm(sG[r][HIDD + j]);
      const float gg = tanhf(sG[r][2 * HIDD + j]);
      const float og = sigm(sG[r][3 * HIDD + j]);
      const float cn = fg * creg[ii] + ig * gg;
      creg[ii] = cn;
      sH[r][j] = og * tanhf(cn);
    }
    // retire prefetch, publish h and the new x buffer
    wait_async0();
    __syncthreads();
  }

  // ---- fused output projection: out = h_last @ W_out^T + b_out ----
  if (wave < 4) {                        // 4 N-tiles cover N = 64
    const int n = wave * 16 + lm;
    v8f acc;
    const float bias = b_out[n];
#pragma unroll
    for (int i = 0; i < 8; ++i) acc[i] = bias;
#pragma unroll
    for (int k0 = 0; k0 < HIDD; k0 += 4) {
      const int ks = k0 + 2 * lh;
      v2f a, b;
      a.x = sH[lm][ks];           a.y = sH[lm][ks + 1];
      b.x = W_out[n * HIDD + ks]; b.y = W_out[n * HIDD + ks + 1];
      acc = wmma4(a, b, acc);
    }
#pragma unroll
    for (int i = 0; i < 8; ++i)
      out[(size_t)(row0 + lh * 8 + i) * OUTD + n] = acc[i];
  }
}

// =====================================================================
extern "C" void kernel_launch(void* const* d_in, const int* in_sizes, int n_in,
                              void* d_out, int out_size, void* d_ws, size_t ws_size,
                              hipStream_t stream) {
  const float* seqs  = (const float*)d_in[0];
  const float* etys  = (const float*)d_in[1];
  const int*   masks = (const int*)  d_in[2];
  const float* W_fus = (const float*)d_in[3];
  const float* b_fus = (const float*)d_in[4];
  const float* Wk    = (const float*)d_in[5];
  const float* bk    = (const float*)d_in[6];
  const float* Wq    = (const float*)d_in[7];
  const float* bq    = (const float*)d_in[8];
  const float* Wv    = (const float*)d_in[9];
  const float* bv    = (const float*)d_in[10];
  const float* W_ih  = (const float*)d_in[11];
  const float* W_hh  = (const float*)d_in[12];
  const float* b_ih  = (const float*)d_in[13];
  const float* b_hh  = (const float*)d_in[14];
  const float* W_out = (const float*)d_in[15];
  const float* b_out = (const float*)d_in[16];

  float* ctxb = (float*)d_ws;    // SEQ*BS*64 floats = 25 MB scratch

  attn_ctx_kernel<<<(BSZ * SEQL) / NPAIR, 128, 0, stream>>>(
      seqs, etys, masks, W_fus, b_fus, Wk, bk, Wq, bq, Wv, bv, ctxb);

  lstm_out_kernel<<<BSZ / 16, 256, 0, stream>>>(
      ctxb, W_ih, W_hh, b_ih, b_hh, W_out, b_out, (float*)d_out);
}